// GCN_20469814132750
// MI455X (gfx1250) — compile-verified
//
#include <hip/hip_runtime.h>
#include <hip/hip_bf16.h>

typedef __attribute__((ext_vector_type(2))) float v2f;
typedef __attribute__((ext_vector_type(8))) float v8f;

// ---------------------------------------------------------------------------
// Zero the atomic accumulators (must run every launch: harness poisons ws once
// and atomics accumulate).
__global__ __launch_bounds__(256) void zero_kernel(float* __restrict__ p, long long n) {
  long long i = (long long)blockIdx.x * blockDim.x + threadIdx.x;
  long long stride = (long long)gridDim.x * blockDim.x;
  for (; i < n; i += stride) p[i] = 0.0f;
}

// deg[v] = in-degree (unweighted, per reference: segment_sum(ones))
__global__ __launch_bounds__(256) void deg_kernel(const int* __restrict__ dst,
                                                  float* __restrict__ deg, int E) {
  int e = blockIdx.x * 256 + threadIdx.x;
  if (e < E) atomicAdd(&deg[dst[e]], 1.0f);
}

// s[dst] += w_e * x[src], D floats per row. D/4 threads per edge, float4 loads,
// 4 global fp32 atomics each. Consecutive lanes cover consecutive chunks of one
// source row -> coalesced gathers, atomic scatters hit L2 (192MB holds h1).
template <int D>
__global__ __launch_bounds__(256) void agg_kernel(const float* __restrict__ x,
                                                  const int* __restrict__ src,
                                                  const int* __restrict__ dst,
                                                  const float* __restrict__ w,
                                                  float* __restrict__ s, int E) {
  const int per = D / 4;
  long long gid = (long long)blockIdx.x * 256 + threadIdx.x;
  int e = (int)(gid / per);
  int c = (int)(gid % per);
  if (e >= E) return;
  int sn = src[e], dn = dst[e];
  float we = w[e];
  float4 xv = ((const float4*)(x + (long long)sn * D))[c];
  float* o = s + (long long)dn * D + c * 4;
  atomicAdd(o + 0, xv.x * we);
  atomicAdd(o + 1, xv.y * we);
  atomicAdd(o + 2, xv.z * we);
  atomicAdd(o + 3, xv.w * we);
}

// in-place h_neigh = s / max(deg, 1)
template <int D>
__global__ __launch_bounds__(256) void div_kernel(float* __restrict__ s,
                                                  const float* __restrict__ deg, int M) {
  long long i = (long long)blockIdx.x * 256 + threadIdx.x;
  if (i >= (long long)M * D) return;
  int v = (int)(i / D);
  s[i] *= 1.0f / fmaxf(deg[v], 1.0f);
}

// ---------------------------------------------------------------------------
// f32 WMMA 16x16x4 helper (ISA layouts, CDNA5 ISA):
//   A 16x4:  v0={K=k+0 lanes0-15, k+2 lanes16-31}, v1={k+1, k+3}, M=lane%16
//   B 4x16:  v0={row k+0 lanes0-15, row k+2 lanes16-31}, v1={k+1,k+3}, N=lane%16
//   C 16x16: vr: lanes0-15 -> M=r, lanes16-31 -> M=r+8, N=lane%16
__device__ __forceinline__ v8f wmma4(v2f a, v2f b, v8f c) {
  return __builtin_amdgcn_wmma_f32_16x16x4_f32(false, a, false, b, (short)0, c,
                                               false, false);
}

// One K-pass over a single [*,Ks] x [Ks,N] operand pair with 2x2 register
// blocking: 4 WMMAs per 4-K chunk, 8 floats loaded per lane per chunk.
// Ks and N are compile-time -> immediate address offsets, no u64 muls.
template <int N, int Ks>
__device__ __forceinline__ void kpass(const float* __restrict__ A,
                                      const float* __restrict__ B,
                                      int ar0, int ar1, int bc0, int bc1,
                                      int lhalf, v8f& c00, v8f& c01, v8f& c10,
                                      v8f& c11) {
  const float* ap0 = A + (size_t)ar0 * Ks + 2 * lhalf;
  const float* ap1 = A + (size_t)ar1 * Ks + 2 * lhalf;
  const float* bp0 = B + (size_t)(2 * lhalf) * N + bc0;
  const float* bp1 = B + (size_t)(2 * lhalf) * N + bc1;
#pragma unroll 4
  for (int k = 0; k < Ks; k += 4) {
    v2f a0, a1, b0, b1;
    a0.x = ap0[k];     a0.y = ap0[k + 1];
    a1.x = ap1[k];     a1.y = ap1[k + 1];
    b0.x = bp0[(size_t)k * N]; b0.y = bp0[(size_t)(k + 1) * N];
    b1.x = bp1[(size_t)k * N]; b1.y = bp1[(size_t)(k + 1) * N];
    c00 = wmma4(a0, b0, c00);
    c01 = wmma4(a0, b1, c01);
    c10 = wmma4(a1, b0, c10);
    c11 = wmma4(a1, b1, c11);
  }
}

template <int N, bool RELU, bool GUARD>
__device__ __forceinline__ void store_tile(float* __restrict__ C, int rt, int bc,
                                           const v8f& acc, float bv, int lhalf,
                                           int M) {
  float* base = C + (size_t)(rt + 8 * lhalf) * N + bc;
#pragma unroll
  for (int r = 0; r < 8; ++r) {
    if (!GUARD || (rt + r + 8 * lhalf) < M) {
      float v = acc[r] + bv;
      if (RELU) v = fmaxf(v, 0.0f);
      base[(size_t)r * N] = v;
    }
  }
}

// SAGE combine as one concatenated GEMM:
//   C[M,N] = [A0 | A1][M, 2*Ks] @ [B0 ; B1][2*Ks, N] + bias (optional ReLU)
// Block = 256 threads (8 waves) -> 64x128 output tile; each wave owns a 32x32
// tile computed as 2x2 WMMA 16x16 tiles. Loads use clamped rows so EXEC stays
// all-1s through the WMMA loops (ISA requirement); stores take an unguarded
// fast path for full tiles.
template <int N, int Ks, bool RELU>
__global__ __launch_bounds__(256) void sage_gemm(const float* __restrict__ A0,
                                                 const float* __restrict__ A1,
                                                 const float* __restrict__ B0,
                                                 const float* __restrict__ B1,
                                                 const float* __restrict__ bias,
                                                 float* __restrict__ C, int M) {
  const int wave  = threadIdx.x >> 5;
  const int lane  = threadIdx.x & 31;
  const int lmod  = lane & 15;
  const int lhalf = lane >> 4;

  const int rowWave = blockIdx.x * 64 + (wave >> 2) * 32;   // 2 row tiles
  const int colWave = blockIdx.y * 128 + (wave & 3) * 32;   // 2 col tiles

  const int r0 = rowWave, r1 = rowWave + 16;
  const int c0 = colWave, c1 = colWave + 16;

  const int ar0 = min(r0 + lmod, M - 1);   // clamp: keeps EXEC full
  const int ar1 = min(r1 + lmod, M - 1);
  const int bc0 = c0 + lmod;
  const int bc1 = c1 + lmod;

  v8f c00 = {}, c01 = {}, c10 = {}, c11 = {};
  kpass<N, Ks>(A0, B0, ar0, ar1, bc0, bc1, lhalf, c00, c01, c10, c11);
  kpass<N, Ks>(A1, B1, ar0, ar1, bc0, bc1, lhalf, c00, c01, c10, c11);

  const float bv0 = bias[bc0];
  const float bv1 = bias[bc1];

  if (rowWave + 32 <= M) {  // full tile: no guards (common path)
    store_tile<N, RELU, false>(C, r0, bc0, c00, bv0, lhalf, M);
    store_tile<N, RELU, false>(C, r0, bc1, c01, bv1, lhalf, M);
    store_tile<N, RELU, false>(C, r1, bc0, c10, bv0, lhalf, M);
    store_tile<N, RELU, false>(C, r1, bc1, c11, bv1, lhalf, M);
  } else {
    store_tile<N, RELU, true>(C, r0, bc0, c00, bv0, lhalf, M);
    store_tile<N, RELU, true>(C, r0, bc1, c01, bv1, lhalf, M);
    store_tile<N, RELU, true>(C, r1, bc0, c10, bv0, lhalf, M);
    store_tile<N, RELU, true>(C, r1, bc1, c11, bv1, lhalf, M);
  }
}

// ---------------------------------------------------------------------------
// Fused MLP head: 128 -> relu 64 -> relu 32 -> 10. 64 threads per node,
// 4 nodes per 256-thread block, staged through LDS.
__global__ __launch_bounds__(256) void mlp_kernel(const float* __restrict__ emb,
                                                  const float* __restrict__ Wl1,
                                                  const float* __restrict__ bl1,
                                                  const float* __restrict__ Wl2,
                                                  const float* __restrict__ bl2,
                                                  const float* __restrict__ Wl3,
                                                  const float* __restrict__ bl3,
                                                  float* __restrict__ out, int M) {
  __shared__ float sbuf[4 * (128 + 64 + 32)];
  const int g = threadIdx.x >> 6;   // node group 0..3
  const int t = threadIdx.x & 63;
  const int node = blockIdx.x * 4 + g;
  float* se  = sbuf + g * 128;
  float* sh1 = sbuf + 512 + g * 64;
  float* sh2 = sbuf + 512 + 256 + g * 32;
  const bool ok = node < M;

  if (ok) {
    se[t]      = emb[(long long)node * 128 + t];
    se[t + 64] = emb[(long long)node * 128 + 64 + t];
  }
  __syncthreads();
  if (ok) {
    float acc = bl1[t];
    for (int k = 0; k < 128; ++k) acc += se[k] * Wl1[k * 64 + t];
    sh1[t] = fmaxf(acc, 0.0f);
  }
  __syncthreads();
  if (ok && t < 32) {
    float acc = bl2[t];
    for (int k = 0; k < 64; ++k) acc += sh1[k] * Wl2[k * 32 + t];
    sh2[t] = fmaxf(acc, 0.0f);
  }
  __syncthreads();
  if (ok && t < 10) {
    float acc = bl3[t];
    for (int k = 0; k < 32; ++k) acc += sh2[k] * Wl3[k * 10 + t];
    out[(long long)node * 10 + t] = acc;
  }
}

// ---------------------------------------------------------------------------
extern "C" void kernel_launch(void* const* d_in, const int* in_sizes, int n_in,
                              void* d_out, int out_size, void* d_ws, size_t ws_size,
                              hipStream_t stream) {
  const float* features = (const float*)d_in[0];
  const int*   esrc     = (const int*)d_in[1];
  const int*   edst     = (const int*)d_in[2];
  const float* ew       = (const float*)d_in[3];
  const float* W1s      = (const float*)d_in[4];
  const float* W1n      = (const float*)d_in[5];
  const float* b1       = (const float*)d_in[6];
  const float* W2s      = (const float*)d_in[7];
  const float* W2n      = (const float*)d_in[8];
  const float* b2       = (const float*)d_in[9];
  const float* Wl1      = (const float*)d_in[10];
  const float* bl1      = (const float*)d_in[11];
  const float* Wl2      = (const float*)d_in[12];
  const float* bl2      = (const float*)d_in[13];
  const float* Wl3      = (const float*)d_in[14];
  const float* bl3      = (const float*)d_in[15];

  const int N = in_sizes[0] / 64;   // 50000 nodes
  const int E = in_sizes[1];        // 800000 edges

  float* out = (float*)d_out;
  float* emb = out + (long long)N * 10;   // tuple member 2, [N,128]

  // Workspace layout (floats): [deg N][s1 N*64][s2 N*256][h1 N*256]
  float* ws  = (float*)d_ws;
  float* deg = ws;
  float* s1  = deg + N;
  float* s2  = s1 + (long long)N * 64;
  float* h1  = s2 + (long long)N * 256;

  // 1) zero deg + s1 + s2 (contiguous)
  long long zn = (long long)N * (1 + 64 + 256);
  zero_kernel<<<4096, 256, 0, stream>>>(ws, zn);

  // 2) degrees
  deg_kernel<<<(E + 255) / 256, 256, 0, stream>>>(edst, deg, E);

  // 3) layer-1 aggregation: s1 = scatter-add(w_e * x[src]), d=64
  {
    long long threads = (long long)E * (64 / 4);
    agg_kernel<64><<<(unsigned)((threads + 255) / 256), 256, 0, stream>>>(
        features, esrc, edst, ew, s1, E);
  }
  // 4) s1 /= max(deg,1)
  div_kernel<64><<<(unsigned)(((long long)N * 64 + 255) / 256), 256, 0, stream>>>(s1, deg, N);

  // 5) h1 = relu([x | s1] @ [W1_self; W1_neigh] + b1)   M=N, N=256, Ks=64
  {
    dim3 grid((N + 63) / 64, 256 / 128);
    sage_gemm<256, 64, true><<<grid, 256, 0, stream>>>(features, s1, W1s, W1n, b1, h1, N);
  }

  // 6) layer-2 aggregation on h1, d=256
  {
    long long threads = (long long)E * (256 / 4);
    agg_kernel<256><<<(unsigned)((threads + 255) / 256), 256, 0, stream>>>(
        h1, esrc, edst, ew, s2, E);
  }
  // 7) s2 /= max(deg,1)
  div_kernel<256><<<(unsigned)(((long long)N * 256 + 255) / 256), 256, 0, stream>>>(s2, deg, N);

  // 8) embeddings = [h1 | s2] @ [W2_self; W2_neigh] + b2  (no relu), N=128, Ks=256
  {
    dim3 grid((N + 63) / 64, 128 / 128);
    sage_gemm<128, 256, false><<<grid, 256, 0, stream>>>(h1, s2, W2s, W2n, b2, emb, N);
  }

  // 9) MLP head -> logits
  mlp_kernel<<<(N + 3) / 4, 256, 0, stream>>>(emb, Wl1, bl1, Wl2, bl2, Wl3, bl3, out, N);
}